// LinearAttention_78297253806115
// MI455X (gfx1250) — compile-verified
//
#include <hip/hip_runtime.h>

typedef __attribute__((ext_vector_type(2))) float v2f;
typedef __attribute__((ext_vector_type(8))) float v8f;

static constexpr int kB = 2;
static constexpr int kC = 64;       // input channels
static constexpr int kH = 4;        // heads
static constexpr int kDH = 32;      // dim head
static constexpr int kHid = 128;    // heads * dim_head
static constexpr int kN = 110592;   // 48^3
static constexpr int kSlices = 128;
static constexpr int kTilesPerSlice = (kN / 16) / kSlices;  // 54

__device__ __forceinline__ v8f wmma_f32_16x16x4(v2f a, v2f b, v8f c) {
  return __builtin_amdgcn_wmma_f32_16x16x4_f32(false, a, false, b, (short)0, c,
                                               false, false);
}

struct WFrag {
  v2f a[16];  // A-operand fragments covering a 16x64 fp32 weight block (K=64)
};

// A-matrix 16x4 f32 layout: VGPR j, lane half hi -> K = 4*kk + 2*hi + j, M = lane&15
__device__ __forceinline__ void load_wfrag(const float* __restrict__ W,
                                           int row0, int ldw, WFrag& f) {
  const int lane = threadIdx.x & 31;
  const int m = lane & 15;
  const int hi = lane >> 4;
  const float* wr = W + (size_t)(row0 + m) * ldw;
#pragma unroll
  for (int kk = 0; kk < 16; ++kk) {
    float2 p = *(const float2*)(wr + kk * 4 + 2 * hi);
    v2f t;
    t[0] = p.x;
    t[1] = p.y;
    f.a[kk] = t;
  }
}

// B-operand fragment from a 64x16 row-major LDS tile (B is KxN = 4x16 per step)
__device__ __forceinline__ v2f bfrag_x(const float* xs, int kk) {
  const int lane = threadIdx.x & 31;
  const int n = lane & 15;
  const int hi = lane >> 4;
  const int r = kk * 4 + 2 * hi;
  v2f t;
  t[0] = xs[r * 16 + n];
  t[1] = xs[(r + 1) * 16 + n];
  return t;
}

__device__ __forceinline__ unsigned lds_offset(const void* p) {
  // generic (flat) LDS addresses carry the LDS byte offset in addr[31:0]
  return (unsigned)(size_t)p;
}

// ---------------------------------------------------------------------------
// Tensor Data Mover: async DMA of one 64-row x 16-col fp32 x-tile into LDS.
// 2D descriptor: tile_dim0=16 (contiguous), tile_dim1=64 rows, row stride = kN.
// Issued by one wave; completion tracked with TENSORcnt (in-order per wave).
// ---------------------------------------------------------------------------
__device__ __forceinline__ void tdm_load_x_tile(const float* __restrict__ gsrc,
                                                unsigned lds_off) {
  typedef __attribute__((ext_vector_type(4))) unsigned int v4u;
  typedef __attribute__((ext_vector_type(8))) int v8i;
  typedef __attribute__((ext_vector_type(4))) int v4i;
  const unsigned long long ga = (unsigned long long)(size_t)gsrc;
  v4u g0;
  g0[0] = 1u;                    // count = 1 (valid user descriptor)
  g0[1] = lds_off;               // lds_addr (bytes)
  g0[2] = (unsigned)ga;          // global_addr[31:0]
  g0[3] = (unsigned)((ga >> 32) & 0x01ffffffu) | (2u << 30);  // addr[56:32] | type=2
  v8i g1;
  g1[0] = (int)(2u << 16);                            // data_size = 4 bytes
  g1[1] = (int)(((unsigned)kN & 0xffffu) << 16);      // tensor_dim0[15:0]
  g1[2] = (int)(((unsigned)kN >> 16) | (64u << 16));  // dim0[31:16] | tensor_dim1=64
  g1[3] = (int)(16u << 16);                           // dim1[31:16]=0 | tile_dim0=16
  g1[4] = (int)64u;                                   // tile_dim1=64 | tile_dim2=0
  g1[5] = (int)kN;                                    // tensor_dim0_stride[31:0]
  g1[6] = 0;                                          // stride0[47:32] | stride1[15:0]
  g1[7] = 0;                                          // stride1[47:16]
  v4i g2 = {0, 0, 0, 0};
  v4i g3 = {0, 0, 0, 0};
  v8i g4 = {0, 0, 0, 0, 0, 0, 0, 0};
  __builtin_amdgcn_tensor_load_to_lds(g0, g1, g2, g3, g4, 0);
}

__device__ __forceinline__ float red_max16(float v) {
#pragma unroll
  for (int off = 8; off > 0; off >>= 1) v = fmaxf(v, __shfl_xor(v, off, 16));
  return v;
}
__device__ __forceinline__ float red_sum16(float v) {
#pragma unroll
  for (int off = 8; off > 0; off >>= 1) v += __shfl_xor(v, off, 16);
  return v;
}

// ---------------------------------------------------------------------------
// Pass 1: per-row max of k = W_k * x over this slice of N.
// grid (kSlices, kB*kH), block 64 (2 waves, one 16-row K tile each)
// ---------------------------------------------------------------------------
__global__ __launch_bounds__(64) void la_kmax_kernel(
    const float* __restrict__ x, const float* __restrict__ w_qkv,
    float* __restrict__ ws_max) {
  const int slice = blockIdx.x;
  const int bh = blockIdx.y;
  const int b = bh / kH;
  const int h = bh % kH;
  const int wave = threadIdx.x >> 5;
  const int lane = threadIdx.x & 31;
  const int hi = lane >> 4;
  __shared__ __align__(16) float xs[2][kC * 16];

  WFrag wk;
  load_wfrag(w_qkv, kHid + h * kDH + wave * 16, kC, wk);
  const float* xb = x + (size_t)b * kC * kN;
  const int tile0 = slice * kTilesPerSlice;

  float rmax[8];
#pragma unroll
  for (int r = 0; r < 8; ++r) rmax[r] = -3.0e38f;

  if (threadIdx.x < 32)
    tdm_load_x_tile(xb + (size_t)tile0 * 16, lds_offset(xs[0]));

  for (int t = 0; t < kTilesPerSlice; ++t) {
    const float* cur = xs[t & 1];
    if (threadIdx.x < 32) {
      if (t + 1 < kTilesPerSlice) {
        tdm_load_x_tile(xb + (size_t)(tile0 + t + 1) * 16,
                        lds_offset(xs[(t + 1) & 1]));
        __builtin_amdgcn_s_wait_tensorcnt(1);
      } else {
        __builtin_amdgcn_s_wait_tensorcnt(0);
      }
    }
    __syncthreads();
    v8f acc = {0.f, 0.f, 0.f, 0.f, 0.f, 0.f, 0.f, 0.f};
#pragma unroll
    for (int kk = 0; kk < 16; ++kk)
      acc = wmma_f32_16x16x4(wk.a[kk], bfrag_x(cur, kk), acc);
#pragma unroll
    for (int r = 0; r < 8; ++r) rmax[r] = fmaxf(rmax[r], acc[r]);
    __syncthreads();  // readers done before this buffer is DMA'd again
  }
#pragma unroll
  for (int r = 0; r < 8; ++r) rmax[r] = red_max16(rmax[r]);
  if ((lane & 15) == 0) {
    float* dst = ws_max + ((size_t)bh * kSlices + slice) * kDH;
#pragma unroll
    for (int r = 0; r < 8; ++r) dst[wave * 16 + r + 8 * hi] = rmax[r];
  }
}

__global__ void la_kmax_reduce(const float* __restrict__ ws_max,
                               float* __restrict__ kmax) {
  const int idx = blockIdx.x * blockDim.x + threadIdx.x;
  if (idx >= kB * kH * kDH) return;
  const int bh = idx / kDH;
  const int c = idx % kDH;
  float m = -3.0e38f;
  for (int s = 0; s < kSlices; ++s)
    m = fmaxf(m, ws_max[((size_t)bh * kSlices + s) * kDH + c]);
  kmax[idx] = m;
}

// ---------------------------------------------------------------------------
// Pass 2: partial ctx = exp(k - m) * v^T and Z = row-sums, per slice.
// grid (kSlices, kB*kH), block 128 (waves 0,1 = K rows; waves 2,3 = V rows;
// then each wave accumulates one 16x16 quadrant of the 32x32 ctx via WMMA)
// ---------------------------------------------------------------------------
__global__ __launch_bounds__(128) void la_ctx_kernel(
    const float* __restrict__ x, const float* __restrict__ w_qkv,
    const float* __restrict__ kmax, float* __restrict__ wsA,
    float* __restrict__ wsZ) {
  const int slice = blockIdx.x;
  const int bh = blockIdx.y;
  const int b = bh / kH;
  const int h = bh % kH;
  const int wave = threadIdx.x >> 5;
  const int lane = threadIdx.x & 31;
  const int m = lane & 15;
  const int hi = lane >> 4;
  __shared__ __align__(16) float xs[2][kC * 16];
  __shared__ float Ps[kDH][18];
  __shared__ float Vs[kDH][18];

  const bool isK = wave < 2;
  const int half = wave & 1;
  const int wrow0 = (isK ? kHid : 2 * kHid) + h * kDH + half * 16;
  WFrag wf;
  load_wfrag(w_qkv, wrow0, kC, wf);

  float mrow[8];
#pragma unroll
  for (int r = 0; r < 8; ++r)
    mrow[r] = isK ? kmax[bh * kDH + half * 16 + r + 8 * hi] : 0.f;

  const float* xb = x + (size_t)b * kC * kN;
  const int tile0 = slice * kTilesPerSlice;
  const int cH = wave >> 1;  // ctx row half
  const int dH = wave & 1;   // ctx col half
  v8f ctx = {0.f, 0.f, 0.f, 0.f, 0.f, 0.f, 0.f, 0.f};
  float zacc[8] = {0.f, 0.f, 0.f, 0.f, 0.f, 0.f, 0.f, 0.f};

  if (threadIdx.x < 32)
    tdm_load_x_tile(xb + (size_t)tile0 * 16, lds_offset(xs[0]));

  for (int t = 0; t < kTilesPerSlice; ++t) {
    const float* cur = xs[t & 1];
    if (threadIdx.x < 32) {
      if (t + 1 < kTilesPerSlice) {
        tdm_load_x_tile(xb + (size_t)(tile0 + t + 1) * 16,
                        lds_offset(xs[(t + 1) & 1]));
        __builtin_amdgcn_s_wait_tensorcnt(1);
      } else {
        __builtin_amdgcn_s_wait_tensorcnt(0);
      }
    }
    __syncthreads();
    v8f kv = {0.f, 0.f, 0.f, 0.f, 0.f, 0.f, 0.f, 0.f};
#pragma unroll
    for (int kk = 0; kk < 16; ++kk)
      kv = wmma_f32_16x16x4(wf.a[kk], bfrag_x(cur, kk), kv);
    if (isK) {
#pragma unroll
      for (int r = 0; r < 8; ++r) {
        const float p = __expf(kv[r] - mrow[r]);
        zacc[r] += p;
        Ps[half * 16 + r + 8 * hi][m] = p;
      }
    } else {
#pragma unroll
      for (int r = 0; r < 8; ++r) Vs[half * 16 + r + 8 * hi][m] = kv[r];
    }
    __syncthreads();
    // ctx quadrant += P(16x16) * V^T(16x16), K = the 16 tile columns
#pragma unroll
    for (int kk = 0; kk < 4; ++kk) {
      const int r0 = kk * 4 + 2 * hi;
      v2f a, bb;
      a[0] = Ps[cH * 16 + m][r0];
      a[1] = Ps[cH * 16 + m][r0 + 1];
      bb[0] = Vs[dH * 16 + m][r0];
      bb[1] = Vs[dH * 16 + m][r0 + 1];
      ctx = wmma_f32_16x16x4(a, bb, ctx);
    }
  }
#pragma unroll
  for (int r = 0; r < 8; ++r) zacc[r] = red_sum16(zacc[r]);
  if (isK && (lane & 15) == 0) {
    float* dz = wsZ + ((size_t)bh * kSlices + slice) * kDH;
#pragma unroll
    for (int r = 0; r < 8; ++r) dz[half * 16 + r + 8 * hi] = zacc[r];
  }
  float* dA = wsA + ((size_t)bh * kSlices + slice) * kDH * kDH;
#pragma unroll
  for (int r = 0; r < 8; ++r)
    dA[(cH * 16 + r + 8 * hi) * kDH + dH * 16 + m] = ctx[r];
}

// ---------------------------------------------------------------------------
// Pass 3: reduce partials, fold W_out and 1/Z into M[b] (64x128).
// M[o][h*32+c] = (1/Z[h,c]) * sum_d W_out[o][h*32+d] * ctx_h[c][d]
// ---------------------------------------------------------------------------
__global__ __launch_bounds__(256) void la_buildM_kernel(
    const float* __restrict__ wsA, const float* __restrict__ wsZ,
    const float* __restrict__ w_out, float* __restrict__ Mmat) {
  const int b = blockIdx.x;
  __shared__ float ctxs[kH * kDH * kDH];
  __shared__ float zinv[kH * kDH];
  for (int e = threadIdx.x; e < kH * kDH * kDH; e += 256) {
    const int h = e / (kDH * kDH);
    const int rem = e % (kDH * kDH);
    const size_t base = ((size_t)(b * kH + h) * kSlices) * kDH * kDH + rem;
    float s = 0.f;
    for (int sl = 0; sl < kSlices; ++sl)
      s += wsA[base + (size_t)sl * kDH * kDH];
    ctxs[e] = s;
  }
  for (int e = threadIdx.x; e < kH * kDH; e += 256) {
    const int h = e / kDH;
    const int c = e % kDH;
    const size_t base = ((size_t)(b * kH + h) * kSlices) * kDH + c;
    float z = 0.f;
    for (int sl = 0; sl < kSlices; ++sl) z += wsZ[base + (size_t)sl * kDH];
    zinv[e] = 1.0f / z;
  }
  __syncthreads();
  for (int e = threadIdx.x; e < kC * kHid; e += 256) {
    const int o = e / kHid;
    const int hc = e % kHid;
    const int h = hc / kDH;
    const int c = hc % kDH;
    const float* wrow = w_out + o * kHid + h * kDH;
    const float* crow = ctxs + (h * kDH + c) * kDH;
    float s = 0.f;
    for (int d = 0; d < kDH; ++d) s += wrow[d] * crow[d];
    Mmat[(size_t)b * kC * kHid + e] = s * zinv[h * kDH + c];
  }
}

// ---------------------------------------------------------------------------
// Pass 4: fused Q = W_q*x, per-column softmax over each head's 32 rows,
// then Y = M * Q + b_out. grid (kSlices, kB), block 256 (8 waves).
// ---------------------------------------------------------------------------
__global__ __launch_bounds__(256) void la_final_kernel(
    const float* __restrict__ x, const float* __restrict__ w_qkv,
    const float* __restrict__ Mmat, const float* __restrict__ b_out,
    float* __restrict__ out) {
  const int slice = blockIdx.x;
  const int b = blockIdx.y;
  const int wave = threadIdx.x >> 5;
  const int lane = threadIdx.x & 31;
  const int m = lane & 15;
  const int hi = lane >> 4;
  __shared__ __align__(16) float xs[2][kC * 16];
  __shared__ float Qs[kHid][17];

  WFrag wq;
  load_wfrag(w_qkv, wave * 16, kC, wq);  // Q block = w_qkv rows 0..127

  const int mwave = wave & 3;  // waves 0..3 own output row tiles
  v2f wm[32];
  {
    const float* mr =
        Mmat + (size_t)b * kC * kHid + (size_t)(mwave * 16 + m) * kHid;
#pragma unroll
    for (int kk = 0; kk < 32; ++kk) {
      float2 p = *(const float2*)(mr + kk * 4 + 2 * hi);
      wm[kk][0] = p.x;
      wm[kk][1] = p.y;
    }
  }
  float bias[8];
#pragma unroll
  for (int r = 0; r < 8; ++r) bias[r] = b_out[mwave * 16 + r + 8 * hi];

  const float* xb = x + (size_t)b * kC * kN;
  float* ob = out + (size_t)b * kC * kN;
  const int tile0 = slice * kTilesPerSlice;

  if (threadIdx.x < 32)
    tdm_load_x_tile(xb + (size_t)tile0 * 16, lds_offset(xs[0]));

  for (int t = 0; t < kTilesPerSlice; ++t) {
    const int i0 = (tile0 + t) * 16;
    const float* cur = xs[t & 1];
    if (threadIdx.x < 32) {
      if (t + 1 < kTilesPerSlice) {
        tdm_load_x_tile(xb + (size_t)(tile0 + t + 1) * 16,
                        lds_offset(xs[(t + 1) & 1]));
        __builtin_amdgcn_s_wait_tensorcnt(1);
      } else {
        __builtin_amdgcn_s_wait_tensorcnt(0);
      }
    }
    __syncthreads();
    // Q tile (16x16) per wave
    v8f q = {0.f, 0.f, 0.f, 0.f, 0.f, 0.f, 0.f, 0.f};
#pragma unroll
    for (int kk = 0; kk < 16; ++kk)
      q = wmma_f32_16x16x4(wq.a[kk], bfrag_x(cur, kk), q);
#pragma unroll
    for (int r = 0; r < 8; ++r) Qs[wave * 16 + r + 8 * hi][m] = q[r];
    __syncthreads();
    // column softmax over each head's 32 rows
    if (threadIdx.x < 64) {
      const int h = threadIdx.x >> 4;
      const int col = threadIdx.x & 15;
      float mx = -3.0e38f;
#pragma unroll
      for (int c = 0; c < kDH; ++c) mx = fmaxf(mx, Qs[h * kDH + c][col]);
      float s = 0.f;
#pragma unroll
      for (int c = 0; c < kDH; ++c) {
        const float e = __expf(Qs[h * kDH + c][col] - mx);
        Qs[h * kDH + c][col] = e;
        s += e;
      }
      const float inv = 1.0f / s;
#pragma unroll
      for (int c = 0; c < kDH; ++c) Qs[h * kDH + c][col] *= inv;
    }
    __syncthreads();
    // Y = M(64x128) * Q(128x16) + b_out ; waves 0..3, K = 128 (32 steps)
    if (wave < 4) {
      v8f y;
#pragma unroll
      for (int r = 0; r < 8; ++r) y[r] = bias[r];
#pragma unroll
      for (int kk = 0; kk < 32; ++kk) {
        const int r0 = kk * 4 + 2 * hi;
        v2f bb;
        bb[0] = Qs[r0][m];
        bb[1] = Qs[r0 + 1][m];
        y = wmma_f32_16x16x4(wm[kk], bb, y);
      }
#pragma unroll
      for (int r = 0; r < 8; ++r)
        ob[(size_t)(wave * 16 + r + 8 * hi) * kN + i0 + m] = y[r];
    }
  }
}

extern "C" void kernel_launch(void* const* d_in, const int* in_sizes, int n_in,
                              void* d_out, int out_size, void* d_ws,
                              size_t ws_size, hipStream_t stream) {
  (void)in_sizes; (void)n_in; (void)out_size; (void)ws_size;
  const float* x = (const float*)d_in[0];
  const float* w_qkv = (const float*)d_in[1];
  const float* w_out = (const float*)d_in[2];
  const float* b_out = (const float*)d_in[3];
  float* out = (float*)d_out;

  float* ws = (float*)d_ws;
  float* ws_max = ws;                                        // 32768 f
  float* kmax = ws_max + (size_t)kB * kH * kSlices * kDH;    // 256 f
  float* wsA = kmax + kB * kH * kDH;                         // 1048576 f
  float* wsZ = wsA + (size_t)kB * kH * kSlices * kDH * kDH;  // 32768 f
  float* Mmat = wsZ + (size_t)kB * kH * kSlices * kDH;       // 16384 f

  la_kmax_kernel<<<dim3(kSlices, kB * kH), 64, 0, stream>>>(x, w_qkv, ws_max);
  la_kmax_reduce<<<1, 256, 0, stream>>>(ws_max, kmax);
  la_ctx_kernel<<<dim3(kSlices, kB * kH), 128, 0, stream>>>(x, w_qkv, kmax,
                                                            wsA, wsZ);
  la_buildM_kernel<<<kB, 256, 0, stream>>>(wsA, wsZ, w_out, Mmat);
  la_final_kernel<<<dim3(kSlices, kB), 256, 0, stream>>>(x, w_qkv, Mmat, b_out,
                                                         out);
}